// JointLoss_66554813219363
// MI455X (gfx1250) — compile-verified
//
#include <hip/hip_runtime.h>
#include <hip/hip_bf16.h>
#include <math.h>

// Problem constants (from reference)
#define BB      32          // batch
#define TT      32          // targets per batch
#define GG      36          // grid
#define AA      5           // anchors
#define NCELL   (BB*GG*GG*AA)        // 207360
#define NLM     136                  // 68*2
#define IMG     288.0f

typedef __attribute__((ext_vector_type(2))) float v2f;
typedef __attribute__((ext_vector_type(8))) float v8f;

// ---------------------------------------------------------------------------
// Wave32 full-sum reduction via V_WMMA_F32_16X16X4_F32.
//   A = {p, 0}:  A[m][0]=p[lane m] (lanes 0-15), A[m][2]=p[lane m+16]; K1/K3 = 0
//   B = ones  :  layout-independent (all elements 1.0)
//   D[m][n] = p[m] + p[m+16]  (constant across n)
// C/D layout: lane L<16 VGPR j = D[j][L]; lane L>=16 VGPR j = D[j+8][L-16].
// Sum of own 8 D regs + xor-16 shuffle = full 32-lane sum (in every lane).
// Requires EXEC == all ones at the call site.
// ---------------------------------------------------------------------------
__device__ __forceinline__ float wave_sum_f32(float p) {
    v2f a; a.x = p;    a.y = 0.0f;
    v2f b; b.x = 1.0f; b.y = 1.0f;
    v8f c = {};
    c = __builtin_amdgcn_wmma_f32_16x16x4_f32(
            /*neg_a=*/false, a, /*neg_b=*/false, b,
            /*c_mod=*/(short)0, c, /*reuse_a=*/false, /*reuse_b=*/false);
    float t = c[0] + c[1] + c[2] + c[3] + c[4] + c[5] + c[6] + c[7];
    t += __shfl_xor(t, 16, 32);
    return t;
}

// ---------------------------------------------------------------------------
// Kernel 1: build compact winner list (sparse scatter resolution).
// One block, 1024 threads: thread = (b, t). Also zeroes counters/accums.
// ws layout: [0..3] int count | [16..47] float accum[8] | [64..] int2 list[1024]
// accum: 0=nme_sum 1=loc_sum 2=sum_all_conf^2 3=obj_se_sum 4=noobj_subtract
// ---------------------------------------------------------------------------
__global__ __launch_bounds__(1024)
void k_targets(const float* __restrict__ bbox_t,
               int* __restrict__ count, float* __restrict__ accum,
               int2* __restrict__ list) {
    const int tid = threadIdx.x;              // 0..1023
    if (tid == 0) *count = 0;
    if (tid < 8)  accum[tid] = 0.0f;

    __shared__ int skey[BB * TT];

    const float* r = bbox_t + tid * 5;
    const float g0 = r[0], g1 = r[1], g2 = r[2], g3 = r[3];
    const bool valid = ((g0 + g1 + g2 + g3 + r[4]) != 0.0f);

    const int gi = (int)(g0 * (float)GG);
    const int gj = (int)(g1 * (float)GG);

    // IoU argmax over anchors (anchors are square: w == h)
    const float gx1 = (g0 - g2 * 0.5f) * IMG, gx2 = (g0 + g2 * 0.5f) * IMG;
    const float gy1 = (g1 - g3 * 0.5f) * IMG, gy2 = (g1 + g3 * 0.5f) * IMG;
    const float acx = (0.5f + (float)gi) / (float)GG;
    const float acy = (0.5f + (float)gj) / (float)GG;
    const float area_g = (gx2 - gx1 + 1.0f) * (gy2 - gy1 + 1.0f);
    const float AW[AA] = {0.24f, 0.12f, 0.08f, 0.28f, 0.15f};

    int best = 0; float best_iou = -3.0e38f;
    #pragma unroll
    for (int a = 0; a < AA; ++a) {
        const float aw = AW[a];
        const float ax1 = (acx - aw * 0.5f) * IMG, ax2 = (acx + aw * 0.5f) * IMG;
        const float ay1 = (acy - aw * 0.5f) * IMG, ay2 = (acy + aw * 0.5f) * IMG;
        const float ix1 = fmaxf(gx1, ax1), iy1 = fmaxf(gy1, ay1);
        const float ix2 = fminf(gx2, ax2), iy2 = fminf(gy2, ay2);
        const float inter  = (ix2 - ix1 + 1.0f) * (iy2 - iy1 + 1.0f);   // may be negative, as in ref
        const float area_a = (ax2 - ax1 + 1.0f) * (ay2 - ay1 + 1.0f);
        const float iou = inter / (area_g + area_a - inter + 1e-16f);
        if (iou > best_iou) { best_iou = iou; best = a; }               // first-max, like argmax
    }

    const int b = tid >> 5;                                            // t = tid & 31
    const bool store = valid && (gi >= 0) && (gi < GG) && (gj >= 0) && (gj < GG); // drop mode
    const int key = store ? (((b * GG + gj) * GG + gi) * AA + best) : -1;
    skey[tid] = key;
    __syncthreads();

    if (key >= 0) {
        // last-write-wins among duplicates within the same batch
        bool win = true;
        const int end = (b << 5) + TT;
        for (int tt = tid + 1; tt < end; ++tt)
            if (skey[tt] == key) { win = false; break; }
        if (win) {
            const int idx = atomicAdd(count, 1);
            list[idx] = make_int2(key, tid);
        }
    }
}

// ---------------------------------------------------------------------------
// Kernel 2: dense Σ conf^2 over all 207,360 cells.
// bbox_prediction is [cell][5]; each thread handles a 20-float (4-cell) chunk.
// conf indices within a chunk: 4,9,14,19 -> float4 vectors 1..4, lanes x,y,z,w
// (vector 0 is never loaded). Wave-reduce via WMMA, one atomic per wave.
// ---------------------------------------------------------------------------
#define NCHUNK (NCELL / 4)   // 51840 chunks of 4 cells
__global__ void k_conf(const float* __restrict__ bbox_p, float* __restrict__ accum) {
    const int chunk = blockIdx.x * blockDim.x + threadIdx.x;
    float p = 0.0f;
    if (chunk < NCHUNK) {
        const float4* v = ((const float4*)bbox_p) + (size_t)chunk * 5;
        const float4 v1 = v[1], v2 = v[2], v3 = v[3], v4 = v[4];
        p = v1.x * v1.x + v2.y * v2.y + v3.z * v3.z + v4.w * v4.w;
    }
    const float tot = wave_sum_f32(p);        // EXEC all ones here (reconverged)
    if ((threadIdx.x & 31) == 0) atomicAdd(&accum[2], tot);
}

// ---------------------------------------------------------------------------
// Kernel 3: per-winner terms. One 128-thread block per possible winner.
// Threads 0..67 gather one landmark each; WMMA wave-sums + shared combine.
// Thread 0 adds: nme term, smooth-L1 loc term, conf obj/noobj corrections.
// ---------------------------------------------------------------------------
__global__ __launch_bounds__(128)
void k_winners(const float* __restrict__ bbox_p, const float* __restrict__ lm_p,
               const float* __restrict__ bbox_t, const float* __restrict__ lm_t,
               const int* __restrict__ count, const int2* __restrict__ list,
               float* __restrict__ accum) {
    const int i = blockIdx.x;
    if (i >= *count) return;                  // uniform per block

    const int2 e   = list[i];
    const int cell  = e.x;
    const int tflat = e.y;

    const float* lp = lm_p + (size_t)cell  * NLM;
    const float* lt = lm_t + (size_t)tflat * NLM;

    const int k = threadIdx.x;
    float d = 0.0f;
    if (k < 68) {
        const float dx = lt[2 * k]     - lp[2 * k];
        const float dy = lt[2 * k + 1] - lp[2 * k + 1];
        d = sqrtf(dx * dx + dy * dy);
    }

    __shared__ float wsum[4];
    const float w = wave_sum_f32(d);          // full-EXEC, per wave
    if ((threadIdx.x & 31) == 0) wsum[threadIdx.x >> 5] = w;
    __syncthreads();

    if (threadIdx.x == 0) {
        const float s = wsum[0] + wsum[1] + wsum[2] + wsum[3];

        const float* g = bbox_t + tflat * 5;
        const float l0 = log1pf(g[0]), l1 = log1pf(g[1]);
        const float l2 = log1pf(g[2]), l3 = log1pf(g[3]);
        const float nf = sqrtf(l2 * l3);      // obj>0 branch of the where()
        atomicAdd(&accum[0], s / nf);

        const float* pb = bbox_p + (size_t)cell * 5;
        const float lg[4] = {l0, l1, l2, l3};
        float loc = 0.0f;
        #pragma unroll
        for (int j = 0; j < 4; ++j) {
            const float df = pb[j] - lg[j];
            const float ad = fabsf(df);
            loc += (ad < 1.0f) ? (0.5f * df * df) : (ad - 0.5f);
        }
        atomicAdd(&accum[1], loc);

        const float cf = pb[4];
        atomicAdd(&accum[3], (cf - 1.0f) * (cf - 1.0f));   // se*obj
        atomicAdd(&accum[4], cf * cf);                     // remove from noobj sum
    }
}

// ---------------------------------------------------------------------------
// Kernel 4: finalize the three scalars.
// ---------------------------------------------------------------------------
__global__ void k_finalize(const int* __restrict__ count,
                           const float* __restrict__ accum,
                           float* __restrict__ out) {
    if (threadIdx.x == 0 && blockIdx.x == 0) {
        const int   c       = *count;
        const float n_obj   = fmaxf((float)c, 1.0f);
        const float n_noobj = fmaxf((float)(NCELL - c), 1.0f);
        out[0] = 2.0f * accum[0] / (68.0f * n_obj);                 // nme (LAMBDA_LM=2)
        out[1] = 5.0f * accum[1] / (n_obj * 4.0f);                  // loc (LAMBDA_COOR=5)
        out[2] = 0.5f * (accum[2] - accum[4]) / n_noobj             // conf (LAMBDA_NOOBJ=0.5)
               + accum[3] / n_obj;
    }
}

// ---------------------------------------------------------------------------
extern "C" void kernel_launch(void* const* d_in, const int* in_sizes, int n_in,
                              void* d_out, int out_size, void* d_ws, size_t ws_size,
                              hipStream_t stream) {
    const float* bbox_p = (const float*)d_in[0];   // (32,36,36,5,5)
    const float* lm_p   = (const float*)d_in[1];   // (32,36,36,5,68,2)
    const float* bbox_t = (const float*)d_in[2];   // (32,32,5)
    const float* lm_t   = (const float*)d_in[3];   // (32,32,68,2)
    float* out = (float*)d_out;                    // (nme, loc_loss, conf_loss)

    int*   d_count = (int*)d_ws;
    float* d_accum = (float*)((char*)d_ws + 16);
    int2*  d_list  = (int2*)((char*)d_ws + 64);

    // 1) sparse scatter resolution + accumulator init (one WGP, 32 waves)
    k_targets<<<1, BB * TT, 0, stream>>>(bbox_t, d_count, d_accum, d_list);

    // 2) dense conf^2 sum: 51840 chunks, 256 threads/block
    const int blocks2 = (NCHUNK + 255) / 256;
    k_conf<<<blocks2, 256, 0, stream>>>(bbox_p, d_accum);

    // 3) per-winner gathers (max 1024 winners)
    k_winners<<<BB * TT, 128, 0, stream>>>(bbox_p, lm_p, bbox_t, lm_t,
                                           d_count, d_list, d_accum);

    // 4) finalize
    k_finalize<<<1, 32, 0, stream>>>(d_count, d_accum, out);
}